// GNNDecoder_20229295964576
// MI455X (gfx1250) — compile-verified
//
#include <hip/hip_runtime.h>
#include <cstdint>

typedef __attribute__((ext_vector_type(2))) float v2f;
typedef __attribute__((ext_vector_type(8))) float v8f;

#define BN_EPS 1e-5f

// ---------------- degree / normalization ----------------
__global__ void k_deg_init(float* __restrict__ deg, int N) {
  int i = blockIdx.x * blockDim.x + threadIdx.x;
  if (i < N) deg[i] = 1.0f;  // self-loop
}

__global__ void k_deg_count(const int* __restrict__ dst, float* __restrict__ deg, int E) {
  int e = blockIdx.x * blockDim.x + threadIdx.x;
  if (e < E) atomicAdd(&deg[dst[e]], 1.0f);
}

__global__ void k_deg_to_dinv(float* __restrict__ deg, int N) {
  int i = blockIdx.x * blockDim.x + threadIdx.x;
  if (i < N) deg[i] = 1.0f / sqrtf(deg[i]);  // deg >= 1 always
}

// ---------------- GEMM1: t1 = x @ W1  (M x 128) * (128 x 128) ----------------
// grid.x = M/16 blocks of 256 threads (8 waves); wave w owns N-tile w (16 cols).
__global__ void k_gemm1(const float* __restrict__ A, const float* __restrict__ B,
                        float* __restrict__ C, int M) {
  __shared__ float sA[16][132];  // pad 132 -> conflict-free column reads
  int mbase = blockIdx.x * 16;
  for (int i = threadIdx.x; i < 16 * 128; i += 256) {
    int r = i >> 7, c = i & 127;
    int row = mbase + r;
    sA[r][c] = (row < M) ? A[(size_t)row * 128 + c] : 0.0f;
  }
  __syncthreads();

  int lane = threadIdx.x & 31;
  int half = lane >> 4;      // 0: K pair {0,1}, 1: K pair {2,3}
  int idx  = lane & 15;      // M for A-frag, N for B-frag
  int col0 = (threadIdx.x >> 5) * 16;

  v8f acc = {};
  for (int k = 0; k < 128; k += 4) {
    int kk = k + 2 * half;
    v2f a, b;
    a.x = sA[idx][kk];
    a.y = sA[idx][kk + 1];
    b.x = B[(size_t)kk * 128 + col0 + idx];
    b.y = B[(size_t)(kk + 1) * 128 + col0 + idx];
    acc = __builtin_amdgcn_wmma_f32_16x16x4_f32(false, a, false, b, (short)0, acc,
                                                false, false);
  }
  for (int r = 0; r < 8; ++r) {
    int row = mbase + r + 8 * half;
    if (row < M) C[(size_t)row * 128 + col0 + idx] = acc[r];
  }
}

// ---------------- GEMM2: t2 = relu(bn(agg1)) @ W2  (M x 128)*(128 x 64) ----------------
// BN+ReLU fused into the LDS staging of A. 32 rows/block, 8 waves: 2 Mtiles x 4 Ntiles.
__global__ void k_gemm2(const float* __restrict__ A, const float* __restrict__ scale,
                        const float* __restrict__ shift, const float* __restrict__ B,
                        float* __restrict__ C, int M) {
  __shared__ float sA[32][132];
  int mbase = blockIdx.x * 32;
  for (int i = threadIdx.x; i < 32 * 128; i += 256) {
    int r = i >> 7, c = i & 127;
    int row = mbase + r;
    float v = 0.0f;
    if (row < M) {
      v = A[(size_t)row * 128 + c] * scale[c] + shift[c];
      v = v > 0.0f ? v : 0.0f;  // ReLU
    }
    sA[r][c] = v;
  }
  __syncthreads();

  int wave = threadIdx.x >> 5;
  int lane = threadIdx.x & 31;
  int half = lane >> 4, idx = lane & 15;
  int row0 = (wave >> 2) * 16;
  int col0 = (wave & 3) * 16;

  v8f acc = {};
  for (int k = 0; k < 128; k += 4) {
    int kk = k + 2 * half;
    v2f a, b;
    a.x = sA[row0 + idx][kk];
    a.y = sA[row0 + idx][kk + 1];
    b.x = B[(size_t)kk * 64 + col0 + idx];
    b.y = B[(size_t)(kk + 1) * 64 + col0 + idx];
    acc = __builtin_amdgcn_wmma_f32_16x16x4_f32(false, a, false, b, (short)0, acc,
                                                false, false);
  }
  for (int r = 0; r < 8; ++r) {
    int row = mbase + row0 + r + 8 * half;
    if (row < M) C[(size_t)row * 64 + col0 + idx] = acc[r];
  }
}

// ---------------- aggregation ----------------
// agg[i][c] = t[i][c] * dinv[i]^2 + bias[c]   (self-loop + post-agg bias)
__global__ void k_init_agg(const float* __restrict__ t, const float* __restrict__ dinv,
                           const float* __restrict__ bias, float* __restrict__ agg,
                           long total, int logC) {
  long i = (long)blockIdx.x * blockDim.x + threadIdx.x;
  if (i >= total) return;
  int node = (int)(i >> logC);
  int c = (int)(i & ((1 << logC) - 1));
  float di = dinv[node];
  agg[i] = t[i] * di * di + bias[c];
}

// one wave per edge; 128 channels -> float4 per lane; atomics land in L2 (agg fits in 192MB L2)
__global__ void k_scatter128(const float* __restrict__ t, float* __restrict__ agg,
                             const int* __restrict__ src, const int* __restrict__ dst,
                             const float* __restrict__ dinv, int E) {
  int eg = blockIdx.x * 8 + (threadIdx.x >> 5);
  if (eg >= E) return;
  int lane = threadIdx.x & 31;
  int s = src[eg], d = dst[eg];
  float nrm = dinv[s] * dinv[d];
  const float4* ts = (const float4*)(t + (size_t)s * 128);
  float4 v = ts[lane];
  float* ad = agg + (size_t)d * 128 + lane * 4;
  atomicAdd(ad + 0, v.x * nrm);
  atomicAdd(ad + 1, v.y * nrm);
  atomicAdd(ad + 2, v.z * nrm);
  atomicAdd(ad + 3, v.w * nrm);
}

__global__ void k_scatter64(const float* __restrict__ t, float* __restrict__ agg,
                            const int* __restrict__ src, const int* __restrict__ dst,
                            const float* __restrict__ dinv, int E) {
  int eg = blockIdx.x * 8 + (threadIdx.x >> 5);
  if (eg >= E) return;
  int lane = threadIdx.x & 31;
  int s = src[eg], d = dst[eg];
  float nrm = dinv[s] * dinv[d];
  const float2* ts = (const float2*)(t + (size_t)s * 64);
  float2 v = ts[lane];
  float* ad = agg + (size_t)d * 64 + lane * 2;
  atomicAdd(ad + 0, v.x * nrm);
  atomicAdd(ad + 1, v.y * nrm);
}

// ---------------- batchnorm ----------------
__global__ void k_zero(float* __restrict__ p, int n) {
  int i = blockIdx.x * blockDim.x + threadIdx.x;
  if (i < n) p[i] = 0.0f;
}

// blockDim.x == C (channel per thread); each block reduces a row chunk, atomic-accumulates.
__global__ void k_bn_partial(const float* __restrict__ x, float* __restrict__ sum,
                             float* __restrict__ sumsq, int N, int rowsPer) {
  int c = threadIdx.x;
  int C = blockDim.x;
  int r0 = blockIdx.x * rowsPer;
  int r1 = r0 + rowsPer; if (r1 > N) r1 = N;
  float s = 0.0f, q = 0.0f;
  for (int r = r0; r < r1; ++r) {
    float v = x[(size_t)r * C + c];
    s += v;
    q += v * v;
  }
  atomicAdd(&sum[c], s);
  atomicAdd(&sumsq[c], q);
}

__global__ void k_bn_final(const float* __restrict__ sum, const float* __restrict__ sumsq,
                           const float* __restrict__ gamma, const float* __restrict__ beta,
                           float* __restrict__ scale, float* __restrict__ shift,
                           int N, int C) {
  int c = blockIdx.x * blockDim.x + threadIdx.x;
  if (c >= C) return;
  float inv_n = 1.0f / (float)N;
  float mu = sum[c] * inv_n;
  float var = sumsq[c] * inv_n - mu * mu;  // biased variance
  if (var < 0.0f) var = 0.0f;
  float rstd = 1.0f / sqrtf(var + BN_EPS);
  float sc = rstd * gamma[c];
  scale[c] = sc;
  shift[c] = beta[c] - mu * sc;
}

__global__ void k_apply_out(const float* __restrict__ agg, const float* __restrict__ scale,
                            const float* __restrict__ shift, float* __restrict__ out,
                            long total) {
  long i = (long)blockIdx.x * blockDim.x + threadIdx.x;
  if (i >= total) return;
  int c = (int)(i & 63);
  out[i] = agg[i] * scale[c] + shift[c];
}

// ---------------- launch ----------------
extern "C" void kernel_launch(void* const* d_in, const int* in_sizes, int n_in,
                              void* d_out, int out_size, void* d_ws, size_t ws_size,
                              hipStream_t stream) {
  const float* x   = (const float*)d_in[0];
  const int*   ei  = (const int*)d_in[1];
  const float* W1  = (const float*)d_in[2];
  const float* b1  = (const float*)d_in[3];
  const float* g1  = (const float*)d_in[4];
  const float* be1 = (const float*)d_in[5];
  const float* W2  = (const float*)d_in[6];
  const float* b2  = (const float*)d_in[7];
  const float* g2  = (const float*)d_in[8];
  const float* be2 = (const float*)d_in[9];

  int N = in_sizes[0] / 128;  // 100000
  int E = in_sizes[1] / 2;    // 1600000
  const int* src = ei;
  const int* dst = ei + E;

  // workspace layout (t2 aliases t1, agg2 aliases agg1 — lifetimes are disjoint)
  char* ws = (char*)d_ws;
  size_t offA = 0;
  size_t offB = (size_t)N * 128 * sizeof(float);
  size_t offD = offB + (size_t)N * 128 * sizeof(float);
  size_t offS = (offD + (size_t)N * sizeof(float) + 255) & ~(size_t)255;
  float* t1   = (float*)(ws + offA);
  float* agg1 = (float*)(ws + offB);
  float* t2   = t1;
  float* agg2 = agg1;
  float* dinv = (float*)(ws + offD);
  float* st   = (float*)(ws + offS);
  float *sum1 = st,        *sq1 = st + 128, *sc1 = st + 256, *sh1 = st + 384;
  float *sum2 = st + 512,  *sq2 = st + 576, *sc2 = st + 640, *sh2 = st + 704;

  // 1) symmetric-norm coefficients
  k_deg_init<<<(N + 255) / 256, 256, 0, stream>>>(dinv, N);
  k_deg_count<<<(E + 255) / 256, 256, 0, stream>>>(dst, dinv, E);
  k_deg_to_dinv<<<(N + 255) / 256, 256, 0, stream>>>(dinv, N);

  // 2) layer 1 linear (fp32 WMMA)
  k_gemm1<<<(N + 15) / 16, 256, 0, stream>>>(x, W1, t1, N);

  // 3) layer 1 aggregation (self-loop init + edge scatter)
  long tot1 = (long)N * 128;
  k_init_agg<<<(int)((tot1 + 255) / 256), 256, 0, stream>>>(t1, dinv, b1, agg1, tot1, 7);
  k_scatter128<<<(E + 7) / 8, 256, 0, stream>>>(t1, agg1, src, dst, dinv, E);

  // 4) BN1 stats -> scale/shift (apply fused into GEMM2)
  k_zero<<<3, 256, 0, stream>>>(st, 768);
  const int rowsPer = 128;
  k_bn_partial<<<(N + rowsPer - 1) / rowsPer, 128, 0, stream>>>(agg1, sum1, sq1, N, rowsPer);
  k_bn_final<<<1, 128, 0, stream>>>(sum1, sq1, g1, be1, sc1, sh1, N, 128);

  // 5) layer 2 linear with fused BN+ReLU on input (fp32 WMMA)
  k_gemm2<<<(N + 31) / 32, 256, 0, stream>>>(agg1, sc1, sh1, W2, t2, N);

  // 6) layer 2 aggregation
  long tot2 = (long)N * 64;
  k_init_agg<<<(int)((tot2 + 255) / 256), 256, 0, stream>>>(t2, dinv, b2, agg2, tot2, 6);
  k_scatter64<<<(E + 7) / 8, 256, 0, stream>>>(t2, agg2, src, dst, dinv, E);

  // 7) BN2 -> final output
  k_bn_partial<<<(N + rowsPer - 1) / rowsPer, 64, 0, stream>>>(agg2, sum2, sq2, N, rowsPer);
  k_bn_final<<<1, 64, 0, stream>>>(sum2, sq2, g2, be2, sc2, sh2, N, 64);
  k_apply_out<<<(int)((tot2 + 255) / 256), 256, 0, stream>>>(agg2, sc2, sh2, (float*)d_out, tot2);
}